// YOLO_15994458210570
// MI455X (gfx1250) — compile-verified
//
#include <hip/hip_runtime.h>
#include <hip/hip_bf16.h>
#include <stdint.h>

// ---- problem constants (match reference) -----------------------------------
#define BATCH   16
#define ANCH    22743
#define NC      80
#define DIMS    85
#define PRE     1000
#define MAXDET  300
#define CONF_THR 0.2f
#define NMS_THR  0.45f
// ---- algorithm constants ---------------------------------------------------
#define NBINS   4096     // radix-select histogram bins over (0.2, 1.0]
#define CAP     4096     // per-batch candidate compaction capacity
#define TILE    256      // anchors staged to LDS per block in the scan passes

typedef unsigned int  u32;
typedef unsigned long long u64;

// ---------------------------------------------------------------------------
// Stage a TILE x 340B anchor tile from global into LDS with coalesced
// gfx1250 async b128 copies.  Tile start is only 4B aligned (340B rows), so
// align down to 16B and carry the skew; preds total size is 16B-multiple so
// aligned chunks never cross the buffer end.
// ---------------------------------------------------------------------------
__device__ __forceinline__
const float* stage_tile(const float* __restrict__ preds, int b, int a0,
                        int nanch, char* tile) {
  size_t byte0 = (((size_t)b * ANCH + a0) * DIMS) * 4;
  u32    skew  = (u32)(byte0 & 15);
  const char* gsrc = (const char*)preds + (byte0 - skew);     // 16B aligned
  size_t bytes = (size_t)nanch * DIMS * 4 + skew;
  int nchunk = (int)((bytes + 15) >> 4);
  for (int i = threadIdx.x; i < nchunk; i += blockDim.x) {
    u32 lds_off = (u32)(uintptr_t)(tile + (size_t)i * 16);    // LDS byte addr
    u64 gaddr   = (u64)(uintptr_t)(gsrc + (size_t)i * 16);
    asm volatile("global_load_async_to_lds_b128 %0, %1, off"
                 :: "v"(lds_off), "v"(gaddr) : "memory");
  }
  asm volatile("s_wait_asynccnt 0" ::: "memory");
  __syncthreads();
  return (const float*)(tile + skew);
}

// ---------------------------------------------------------------------------
// Kernel 0: zero workspace counters/histogram
// ---------------------------------------------------------------------------
__global__ void k_init(u32* __restrict__ hist, u32* __restrict__ cnt) {
  int i = blockIdx.x * blockDim.x + threadIdx.x;
  if (i < BATCH * NBINS) hist[i] = 0u;
  if (i < BATCH)         cnt[i]  = 0u;
}

// ---------------------------------------------------------------------------
// Kernel 1: per-batch score histogram.  First full pass over preds (123.7MB,
// HBM-bound ~5.3us @ 23.3TB/s).  Global reads fully coalesced via async
// staging; LDS row stride 85 dwords is bank-conflict-free (gcd(85,64)=1).
// ---------------------------------------------------------------------------
__global__ void k_hist(const float* __restrict__ preds, u32* __restrict__ hist) {
  __shared__ u32 lh[NBINS];
  extern __shared__ __align__(16) char tile[];          // TILE*340 + 16 bytes
  for (int i = threadIdx.x; i < NBINS; i += blockDim.x) lh[i] = 0u;
  __syncthreads();

  const int b  = blockIdx.y;
  const int a0 = blockIdx.x * TILE;
  int nanch = ANCH - a0; if (nanch > TILE) nanch = TILE;
  const float* base = stage_tile(preds, b, a0, nanch, tile);

  if ((int)threadIdx.x < nanch) {
    const float* row = base + (size_t)threadIdx.x * DIMS;   // LDS
    const float conf = row[4];
    #pragma unroll 4
    for (int c = 0; c < NC; ++c) {
      float s = row[5 + c] * conf;
      if (s > CONF_THR) {
        int bin = (int)((s - CONF_THR) * ((float)NBINS / 0.8f));
        bin = bin < 0 ? 0 : (bin > NBINS - 1 ? NBINS - 1 : bin);
        atomicAdd(&lh[bin], 1u);
      }
    }
  }
  __syncthreads();
  for (int i = threadIdx.x; i < NBINS; i += blockDim.x)
    if (lh[i]) atomicAdd(&hist[b * NBINS + i], lh[i]);
}

// ---------------------------------------------------------------------------
// Kernel 2: find per-batch threshold bin containing the rank-PRE score
// ---------------------------------------------------------------------------
__global__ void k_thresh(const u32* __restrict__ hist, int* __restrict__ tbin) {
  int b = threadIdx.x;
  if (b >= BATCH) return;
  u32 acc = 0; int t = 0;
  for (int i = NBINS - 1; i >= 0; --i) {
    acc += hist[b * NBINS + i];
    if (acc >= PRE) { t = i; break; }
  }
  tbin[b] = t;
}

// ---------------------------------------------------------------------------
// Kernel 3: compaction pass.  Second read of preds is L2-resident
// (123.7MB < 192MB L2).  Same async-staged access pattern.
// packed key = (score_bits << 32) | ~flat_idx  (desc sort => top_k order)
// ---------------------------------------------------------------------------
__global__ void k_compact(const float* __restrict__ preds,
                          const int* __restrict__ tbin,
                          u32* __restrict__ cnt,
                          u64* __restrict__ cand) {
  extern __shared__ __align__(16) char tile[];
  const int b  = blockIdx.y;
  const int a0 = blockIdx.x * TILE;
  int nanch = ANCH - a0; if (nanch > TILE) nanch = TILE;
  const float* base = stage_tile(preds, b, a0, nanch, tile);

  if ((int)threadIdx.x >= nanch) return;
  const int t = tbin[b];
  const float* row = base + (size_t)threadIdx.x * DIMS;
  const float conf = row[4];
  const int a = a0 + threadIdx.x;
  #pragma unroll 4
  for (int c = 0; c < NC; ++c) {
    float s = row[5 + c] * conf;
    if (s > CONF_THR) {
      int bin = (int)((s - CONF_THR) * ((float)NBINS / 0.8f));
      bin = bin < 0 ? 0 : (bin > NBINS - 1 ? NBINS - 1 : bin);
      if (bin >= t) {
        u32 pos = atomicAdd(&cnt[b], 1u);
        if (pos < CAP) {
          u32 fidx = (u32)(a * NC + c);
          cand[(size_t)b * CAP + pos] =
              ((u64)__float_as_uint(s) << 32) | (u64)(~fidx);
        }
      }
    }
  }
}

// ---------------------------------------------------------------------------
// Kernel 4: per-batch sort + NMS + output.  One 1024-thread block per batch,
// ~204KB of LDS (CDNA5 320KB/WGP makes the 128KB suppression bit-matrix fit).
// ---------------------------------------------------------------------------
__global__ __launch_bounds__(1024, 1)
void k_nms(const float* __restrict__ preds,
           const u64* __restrict__ cand,
           const u32* __restrict__ cnt,
           float* __restrict__ out) {
  extern __shared__ __align__(16) char smem[];
  u64*   skey  = (u64*)  (smem);            //  32768 B : sort keys
  float* bs    = (float*)(smem + 32768);    //  16384 B : class-shifted boxes
  float* bo    = (float*)(smem + 49152);    //  16384 B : original boxes
  float* sc    = (float*)(smem + 65536);    //   4096 B : scores
  int*   lb    = (int*)  (smem + 69632);    //   4096 B : labels
  float* red   = (float*)(smem + 73728);    //   4096 B : max-reduce, then areas
  u32*   sup   = (u32*)  (smem + 77824);    // 131072 B : 1024x1024 supp bits
  u32*   keepm = (u32*)  (smem + 208896);   //    128 B : final keep mask
  const int b   = blockIdx.x;
  const int tid = threadIdx.x;

  // ---- stage candidates into LDS via async global->LDS mover (ASYNCcnt) ----
  const u64* src = cand + (size_t)b * CAP;
  for (int i = tid; i < CAP / 2; i += 1024) {           // 16B per op, 32KB total
    u32 lds_off = (u32)(uintptr_t)(&skey[2 * i]);
    u64 gaddr   = (u64)(uintptr_t)(src + 2 * i);
    asm volatile("global_load_async_to_lds_b128 %0, %1, off"
                 :: "v"(lds_off), "v"(gaddr) : "memory");
  }
  asm volatile("s_wait_asynccnt 0" ::: "memory");
  __syncthreads();

  int count = (int)cnt[b]; if (count > CAP) count = CAP;
  for (int i = tid; i < CAP; i += 1024)
    if (i >= count) skey[i] = 0ull;                     // pad: score 0.0 < CONF
  __syncthreads();

  // ---- bitonic sort, descending (score desc, idx asc via ~idx) -------------
  for (u32 k = 2; k <= CAP; k <<= 1) {
    for (u32 j = k >> 1; j > 0; j >>= 1) {
      for (u32 t = tid; t < CAP; t += 1024) {
        u32 ixj = t ^ j;
        if (ixj > t) {
          u64 x = skey[t], y = skey[ixj];
          bool up = ((t & k) == 0);
          if (up ? (x < y) : (x > y)) { skey[t] = y; skey[ixj] = x; }
        }
      }
      __syncthreads();
    }
  }

  // ---- extract top-1024, gather boxes, compute class shift -----------------
  {
    const int i = tid;
    u64  key  = skey[i];
    float s   = __uint_as_float((u32)(key >> 32));
    u32  fidx = ~((u32)key);
    bool valid = (s > CONF_THR) && (i < PRE) && (i < count);
    int label  = valid ? (int)(fidx % NC) : 0;
    int anchor = valid ? (int)(fidx / NC) : 0;
    float b0 = 0.f, b1 = 0.f, b2 = 0.f, b3 = 0.f;
    if (valid) {
      const float* p = preds + ((size_t)b * ANCH + anchor) * DIMS;
      b0 = p[0]; b1 = p[1]; b2 = p[2]; b3 = p[3];
    }
    bo[i * 4 + 0] = b0; bo[i * 4 + 1] = b1;
    bo[i * 4 + 2] = b2; bo[i * 4 + 3] = b3;
    sc[i] = valid ? s : -1.0f;
    lb[i] = label;
    // block max over valid box coords (reference: jnp.max(cand_boxes))
    red[i] = valid ? fmaxf(fmaxf(b0, b1), fmaxf(b2, b3)) : -3.0e38f;
    __syncthreads();
    for (int off = 512; off > 0; off >>= 1) {
      if (tid < off) red[tid] = fmaxf(red[tid], red[tid + off]);
      __syncthreads();
    }
    float mx = red[0];
    __syncthreads();
    float shift = (float)label * (mx + 1.0f);
    if (valid) {
      bs[i * 4 + 0] = b0 + shift; bs[i * 4 + 1] = b1 + shift;
      bs[i * 4 + 2] = b2 + shift; bs[i * 4 + 3] = b3 + shift;
    } else {
      float park = -1.0e8f - (float)i * 64.0f;   // zero-area, non-overlapping
      bs[i * 4 + 0] = park; bs[i * 4 + 1] = park;
      bs[i * 4 + 2] = park; bs[i * 4 + 3] = park;
    }
    red[i] = fmaxf(bs[i * 4 + 2] - bs[i * 4 + 0], 0.0f) *
             fmaxf(bs[i * 4 + 3] - bs[i * 4 + 1], 0.0f);   // area (reuse red)
  }
  __syncthreads();

  // ---- suppression bit-matrix: row i, bit j set iff iou(i,j)>TH && j>i -----
  {
    const int i = tid;
    const float x1 = bs[i * 4 + 0], y1 = bs[i * 4 + 1];
    const float x2 = bs[i * 4 + 2], y2 = bs[i * 4 + 3];
    const float ai = red[i];
    for (int w = 0; w < 32; ++w) {
      u32 bits = 0;
      const int jb = w * 32;
      for (int jj = 0; jj < 32; ++jj) {       // LDS broadcast: all lanes same j
        const int j = jb + jj;
        float xx1 = fmaxf(x1, bs[j * 4 + 0]);
        float yy1 = fmaxf(y1, bs[j * 4 + 1]);
        float xx2 = fminf(x2, bs[j * 4 + 2]);
        float yy2 = fminf(y2, bs[j * 4 + 3]);
        float inter = fmaxf(xx2 - xx1, 0.0f) * fmaxf(yy2 - yy1, 0.0f);
        float iou = inter / (ai + red[j] - inter + 1e-7f);
        bits |= ((iou > NMS_THR) && (j > i)) ? (1u << jj) : 0u;
      }
      sup[i * 32 + w] = bits;
    }
  }
  __syncthreads();

  // ---- greedy scan: single wave32, removal mask lives in 32 lane registers -
  if (tid < 32) {
    const int lane = tid;
    u32 valw = 0;
    for (int t2 = 0; t2 < 32; ++t2) {
      int i = lane * 32 + t2;
      if (sc[i] > CONF_THR && i < PRE) valw |= (1u << t2);
    }
    u32 rm = 0;
    for (int i = 0; i < PRE; ++i) {
      int w = i >> 5, bpos = i & 31;
      u32 rw = (u32)__shfl((int)rm,   w, 32);   // wave32 broadcast
      u32 vw = (u32)__shfl((int)valw, w, 32);
      bool alive = ((vw >> bpos) & 1u) && !((rw >> bpos) & 1u);
      if (alive) rm |= sup[i * 32 + lane];
    }
    keepm[lane] = valw & ~rm;
  }
  __syncthreads();

  // ---- outputs: boxes | scores | labels(int32), concatenated ---------------
  float* obox = out;
  float* oscr = out + (size_t)BATCH * MAXDET * 4;
  int*   olab = (int*)(out + (size_t)BATCH * MAXDET * 5);
  if (tid < MAXDET) {                         // default-fill (d_out is poisoned)
    int o = b * MAXDET + tid;
    obox[o * 4 + 0] = 0.f; obox[o * 4 + 1] = 0.f;
    obox[o * 4 + 2] = 0.f; obox[o * 4 + 3] = 0.f;
    oscr[o] = 0.f; olab[o] = -1;
  }
  __syncthreads();
  if (tid < 32) {                             // wave32 ballot compaction
    const int lane = tid; int base = 0;
    for (int c = 0; c < 32 && base < MAXDET; ++c) {
      u32 kw = keepm[c];
      bool kf = (kw >> lane) & 1u;
      u32 mask = (u32)__ballot(kf);
      int pos = base + __popc(mask & ((1u << lane) - 1u));
      if (kf && pos < MAXDET) {
        int i = c * 32 + lane;
        int o = b * MAXDET + pos;
        oscr[o] = sc[i];
        olab[o] = lb[i];
        obox[o * 4 + 0] = bo[i * 4 + 0]; obox[o * 4 + 1] = bo[i * 4 + 1];
        obox[o * 4 + 2] = bo[i * 4 + 2]; obox[o * 4 + 3] = bo[i * 4 + 3];
      }
      base += __popc(mask);
    }
  }
}

// ---------------------------------------------------------------------------
extern "C" void kernel_launch(void* const* d_in, const int* in_sizes, int n_in,
                              void* d_out, int out_size, void* d_ws, size_t ws_size,
                              hipStream_t stream) {
  (void)in_sizes; (void)n_in; (void)out_size; (void)ws_size;
  const float* preds = (const float*)d_in[0];
  float* out = (float*)d_out;
  char* ws = (char*)d_ws;
  // workspace layout (16-byte aligned slices, ~0.77 MB total)
  u32* hist = (u32*)(ws);                                   // 16*4096 u32
  u32* cnt  = (u32*)(ws + (size_t)BATCH * NBINS * 4);       // 16 u32
  int* tbin = (int*)(ws + (size_t)BATCH * NBINS * 4 + 64);  // 16 i32
  u64* cand = (u64*)(ws + (size_t)BATCH * NBINS * 4 + 128); // 16*4096 u64

  k_init<<<dim3((BATCH * NBINS + 255) / 256), dim3(256), 0, stream>>>(hist, cnt);

  dim3 gscan((ANCH + TILE - 1) / TILE, BATCH);
  const size_t tile_smem = (size_t)TILE * DIMS * 4 + 16;    // 87056 B
  k_hist<<<gscan, dim3(TILE), tile_smem, stream>>>(preds, hist);

  k_thresh<<<dim3(1), dim3(32), 0, stream>>>(hist, tbin);

  k_compact<<<gscan, dim3(TILE), tile_smem, stream>>>(preds, tbin, cnt, cand);

  const size_t nms_smem = 209024;   // 204 KB dynamic LDS (CDNA5: 320 KB/WGP)
  k_nms<<<dim3(BATCH), dim3(1024), nms_smem, stream>>>(preds, cand, cnt, out);
}